// MaxwellNeuron_87162066305735
// MI455X (gfx1250) — compile-verified
//
#include <hip/hip_runtime.h>

// Maxwell element scan: stress[t] = a*stress[t-1] + u[t], a = exp(-DT*E/eta)
// Single-pass, one block per row (B=256), 8 waves/block, chunk = 2048 elems.
// Intra-16-step scan via V_WMMA_F32_16X16X4_F32 (scan-as-lower-triangular
// matmul); carries via wave shuffle scan + LDS + serial chunk carry.
// Strain streaming uses GLOBAL_LOAD_ASYNC_TO_LDS_B128 double-buffering
// (ASYNCcnt) when the builtin is available; register double-buffer otherwise.

typedef float v2f __attribute__((ext_vector_type(2)));
typedef float v8f __attribute__((ext_vector_type(8)));
typedef int   v4i __attribute__((vector_size(16)));   // matches builtin proto

#define DT_C     0.1f
#define CHUNK    2048
#define SEGPAD   20            // padded LDS stride per 16-step segment (floats)
#define WAVEPAD  (16*SEGPAD)   // 320 floats of LDS per wave tile

#if defined(__has_builtin)
#if __has_builtin(__builtin_amdgcn_global_load_async_to_lds_b128)
#define USE_ASYNC_LDS 1
#endif
#endif

#define AS1 __attribute__((address_space(1)))
#define AS3 __attribute__((address_space(3)))

#ifdef USE_ASYNC_LDS
static __device__ __forceinline__ AS3 v4i* to_as3(const void* p) {
  // flat LDS pointer low 32 bits == LDS byte offset (ISA 10.2 aperture rules)
  return (AS3 v4i*)(unsigned int)(unsigned long long)p;
}
static __device__ __forceinline__ AS1 v4i* to_as1(const void* p) {
  return (AS1 v4i*)(unsigned long long)p;
}
static __device__ __forceinline__ void wait_async0() {
#if __has_builtin(__builtin_amdgcn_s_wait_asynccnt)
  __builtin_amdgcn_s_wait_asynccnt(0);
#else
  asm volatile("s_wait_asynccnt 0x0" ::: "memory");
#endif
}
#endif

__global__ __launch_bounds__(256)
void maxwell_wmma_scan(const float* __restrict__ strain,
                       const float* __restrict__ rate,
                       const float* __restrict__ logE,
                       const float* __restrict__ logEta,
                       float* __restrict__ out,
                       int S) {
#ifdef USE_ASYNC_LDS
  __shared__ __align__(16) float sBuf[2][CHUNK];    // async double buffer
#else
  __shared__ __align__(16) float sL[CHUNK];         // raw strain chunk
#endif
  __shared__ __align__(16) float uPad[8 * WAVEPAD]; // u tile, reused for stress
  __shared__ float eArr[8];                         // per-wave local tile ends
  __shared__ float Warr[8];                         // per-wave carry-in
  __shared__ float prevScalar[2];                   // last strain of chunk c in slot c&1

  const int t    = threadIdx.x;
  const int lane = t & 31;
  const int wv   = t >> 5;
  const int hi   = (lane >= 16) ? 1 : 0;
  const int mrow = lane & 15;
  const int row  = blockIdx.x;
  const int nch  = S / CHUNK;

  const float E   = expf(logE[0]);
  const float eta = expf(logEta[0]);
  const float lam = DT_C * E / eta;          // a = exp(-lam)

  // ---- per-lane constants (computed once) ----
  const float a256 = expf(-256.0f * lam);
  float a16v[4];                             // a^16, a^32, a^64, a^128
  #pragma unroll
  for (int dd = 0; dd < 4; ++dd) a16v[dd] = expf(-16.0f * lam * (float)(1 << dd));
  const float pw16n = expf(-16.0f * lam * (float)mrow);   // a^(16n), n = lane-16
  float powStep[8];                          // a^(m+1), m = v + 8*hi
  #pragma unroll
  for (int v = 0; v < 8; ++v) powStep[v] = expf(-lam * (float)(v + 8 * hi + 1));

  // A-matrix fragments: L[m][k] = (m>=k) ? a^(m-k) : 0, 16x4 slice per WMMA.
  // Layout (ISA 7.12.2): lanes 0-15 M=0..15 {K=4s+0,4s+1}; lanes 16-31 {K=4s+2,4s+3}.
  v2f aFrag[4];
  #pragma unroll
  for (int s = 0; s < 4; ++s) {
    const int k0 = 4 * s + 2 * hi;
    aFrag[s].x = (mrow >= k0)     ? expf(-lam * (float)(mrow - k0))     : 0.0f;
    aFrag[s].y = (mrow >= k0 + 1) ? expf(-lam * (float)(mrow - k0 - 1)) : 0.0f;
  }

  const size_t rowBase = (size_t)row * (size_t)S;
  // virtual s[-1] so that u[0] = E*(s[0]-virt) = E*s[0] + eta*rate[0]
  const float virt = -(eta / E) * rate[rowBase];

  float cReg = 0.0f;   // chunk carry (only thread 0's copy is meaningful)

#ifdef USE_ASYNC_LDS
  // prologue: async-load chunk 0 into sBuf[0]
  {
    const float* g = strain + rowBase;
    __builtin_amdgcn_global_load_async_to_lds_b128(
        to_as1(g + 4 * t), to_as3(&sBuf[0][4 * t]), 0, 0);
    __builtin_amdgcn_global_load_async_to_lds_b128(
        to_as1(g + 1024 + 4 * t), to_as3(&sBuf[0][1024 + 4 * t]), 0, 0);
  }
#else
  // prologue: register double buffer for chunk 0
  float4 cur0 = *(const float4*)(strain + rowBase + 4 * t);
  float4 cur1 = *(const float4*)(strain + rowBase + 1024 + 4 * t);
#endif

  for (int chunk = 0; chunk < nch; ++chunk) {
    const size_t gbase = rowBase + (size_t)chunk * CHUNK;

#ifdef USE_ASYNC_LDS
    const int b = chunk & 1;
    float* sv = sBuf[b];
    // this chunk's async loads (and any older traffic) must be complete
    wait_async0();
    __syncthreads();
    // prefetch chunk+1 into the other buffer; overlaps all compute below
    if (chunk + 1 < nch) {
      const float* g = strain + gbase + CHUNK;
      __builtin_amdgcn_global_load_async_to_lds_b128(
          to_as1(g + 4 * t), to_as3(&sBuf[b ^ 1][4 * t]), 0, 0);
      __builtin_amdgcn_global_load_async_to_lds_b128(
          to_as1(g + 1024 + 4 * t), to_as3(&sBuf[b ^ 1][1024 + 4 * t]), 0, 0);
    }
#else
    float* sv = sL;
    __syncthreads();   // prev iteration done reading sL / uPad
    *(float4*)&sL[4 * t]        = cur0;
    *(float4*)&sL[1024 + 4 * t] = cur1;
    float4 n0 = make_float4(0.f, 0.f, 0.f, 0.f);
    float4 n1 = make_float4(0.f, 0.f, 0.f, 0.f);
    if (chunk + 1 < nch) {
      n0 = *(const float4*)(strain + gbase + CHUNK + 4 * t);
      n1 = *(const float4*)(strain + gbase + CHUNK + 1024 + 4 * t);
    }
    __syncthreads();
#endif

    // ---- build u into padded LDS tile (reads sv, writes uPad) ----
    #pragma unroll
    for (int k = 0; k < 8; ++k) {
      const int i = t + k * 256;
      float prev;
      if (i == 0) prev = (chunk == 0) ? virt : prevScalar[(chunk & 1) ^ 1];
      else        prev = sv[i - 1];
      const float s_i = sv[i];
      const float uv = E * (s_i - prev);
      const int w = i >> 8, r = i & 255;
      uPad[w * WAVEPAD + (r >> 4) * SEGPAD + (r & 15)] = uv;
      if (i == CHUNK - 1) prevScalar[chunk & 1] = s_i;   // halo for next chunk
    }
    __syncthreads();

    // ---- local 16-step scans of 16 segments via 4 chained f32 WMMAs ----
    v8f d = {0.f, 0.f, 0.f, 0.f, 0.f, 0.f, 0.f, 0.f};
    const int bBase = wv * WAVEPAD + mrow * SEGPAD + 2 * hi;
    #pragma unroll
    for (int s = 0; s < 4; ++s) {
      v2f bf = *(const v2f*)&uPad[bBase + 4 * s];
      d = __builtin_amdgcn_wmma_f32_16x16x4_f32(
              /*neg_a=*/false, aFrag[s],
              /*neg_b=*/false, bf,
              /*c_mod=*/(short)0, d,
              /*reuse_a=*/false, /*reuse_b=*/false);
    }

    // local tile end = D[15][15] = vgpr7 @ lane 31
    const float segEnd = d[7];
    if (lane == 31) eArr[wv] = segEnd;
    __syncthreads();

    // ---- cross-wave carry scan (8 values, serial on thread 0) ----
    if (t == 0) {
      float run = cReg;
      #pragma unroll
      for (int w = 0; w < 8; ++w) { Warr[w] = run; run = a256 * run + eArr[w]; }
      cReg = run;   // carry into next chunk
    }
    __syncthreads();
    const float W = Warr[wv];

    // ---- cross-segment carry scan (lanes 16..31 hold segment ends in d[7]) ----
    float x = segEnd;
    #pragma unroll
    for (int dd = 0; dd < 4; ++dd) {
      const int off = 1 << dd;
      const float y = __shfl_up(x, off, 32);
      if (lane >= 16 + off) x += a16v[dd] * y;
    }
    // carry into segment n: c[n] = a^(16n)*W + incl[n-1]
    const float prevIncl = __shfl_up(x, 1, 32);
    const float Cv   = pw16n * W + ((lane > 16) ? prevIncl : 0.0f);
    const float Cseg = __shfl(Cv, 16 + mrow, 32);

    // ---- apply carries, stage stress back into uPad (own-wave region) ----
    const int dBase = wv * WAVEPAD + mrow * SEGPAD + 8 * hi;
    #pragma unroll
    for (int v = 0; v < 8; ++v) {
      uPad[dBase + v] = d[v] + powStep[v] * Cseg;
    }
    __syncthreads();

    // ---- coalesced float4 store ----
    #pragma unroll
    for (int k = 0; k < 2; ++k) {
      const int i0 = 4 * (t + k * 256);
      const int w = i0 >> 8, r = i0 & 255;
      const float4 val = *(const float4*)&uPad[w * WAVEPAD + (r >> 4) * SEGPAD + (r & 15)];
      *(float4*)(out + gbase + i0) = val;
    }

#ifndef USE_ASYNC_LDS
    cur0 = n0;
    cur1 = n1;
#endif
  }
}

extern "C" void kernel_launch(void* const* d_in, const int* in_sizes, int n_in,
                              void* d_out, int out_size, void* d_ws, size_t ws_size,
                              hipStream_t stream) {
  const float* strain = (const float*)d_in[0];
  const float* rate   = (const float*)d_in[1];
  const float* logE   = (const float*)d_in[2];
  const float* logEta = (const float*)d_in[3];
  float* out = (float*)d_out;

  const int S = 65536;
  const int B = in_sizes[0] / S;

  maxwell_wmma_scan<<<B, 256, 0, stream>>>(strain, rate, logE, logEta, out, S);
}